// MultiHeadGATLayer_28355374088752
// MI455X (gfx1250) — compile-verified
//
#include <hip/hip_runtime.h>
#include <math.h>

#define N_NODES 50000
#define N_EDGES 1600000
#define IN_DIM  128
#define OUT_DIM 32
#define N_HEADS 4
#define TOT_OUT (N_HEADS * OUT_DIM)   // 128

typedef __attribute__((ext_vector_type(2))) float v2f;
typedef __attribute__((ext_vector_type(8))) float v8f;

// ---------------------------------------------------------------------------
// Kernel 0: init accumulation buffers (harness poisons d_out/d_ws with 0xAA)
// ---------------------------------------------------------------------------
__global__ void k_init(float* __restrict__ outh,
                       float* __restrict__ sumb,
                       float* __restrict__ maxb) {
  int i = blockIdx.x * blockDim.x + threadIdx.x;
  if (i < N_NODES * TOT_OUT) outh[i] = 0.0f;
  if (i < N_NODES * N_HEADS) { sumb[i] = 0.0f; maxb[i] = -INFINITY; }
}

// ---------------------------------------------------------------------------
// Kernel 1: Wh[n, h*32+o] = sum_i x[n,i] * weight[h,i,o]
// Exact-f32 GEMM via V_WMMA_F32_16X16X4_F32. One wave per 16x16 tile.
// Block = 256 threads = 8 waves -> the 8 column tiles (128 outputs) of one
// 16-row tile. 50000 = 3125*16, 128 = 8*16, 128 = 32*4: no tails anywhere.
// A 16x4 frag (2 VGPR): lanes 0-15 carry K=k0,k0+1; lanes 16-31 K=k0+2,k0+3.
// D 16x16 (8 VGPR): VGPR r -> M=r (lanes 0-15) / M=8+r (lanes 16-31), N=lane&15.
// ---------------------------------------------------------------------------
__global__ void k_gemm_wmma(const float* __restrict__ x,
                            const float* __restrict__ weight,
                            float* __restrict__ Wh) {
  const int wave  = threadIdx.x >> 5;       // 0..7 -> N tile
  const int lane  = threadIdx.x & 31;
  const int m0    = blockIdx.x * 16;
  const int n0    = wave * 16;

  const int l15   = lane & 15;
  const int khalf = (lane >> 4) * 2;        // 0 or 2

  const int n = n0 + l15;                   // global output column 0..127
  const int h = n >> 5;                     // head
  const int o = n & 31;                     // out-dim within head

  const float* __restrict__ xrow = x + (m0 + l15) * IN_DIM;
  const float* __restrict__ wcol = weight + h * IN_DIM * OUT_DIM + o;

  v8f acc = {};
#pragma unroll 4
  for (int k0 = 0; k0 < IN_DIM; k0 += 4) {
    const int ka = k0 + khalf;
    v2f a, b;
    a.x = xrow[ka];
    a.y = xrow[ka + 1];
    b.x = wcol[ka * OUT_DIM];               // W2[ka,   n]
    b.y = wcol[(ka + 1) * OUT_DIM];         // W2[ka+1, n]
    acc = __builtin_amdgcn_wmma_f32_16x16x4_f32(
        /*neg_a=*/false, a, /*neg_b=*/false, b,
        /*c_mod=*/(short)0, acc, /*reuse_a=*/false, /*reuse_b=*/false);
  }

  const int mbase = m0 + (lane >> 4) * 8;
#pragma unroll
  for (int r = 0; r < 8; ++r)
    Wh[(mbase + r) * TOT_OUT + n0 + l15] = acc[r];
}

// ---------------------------------------------------------------------------
// Kernel 2: per-node attention scores
//   s_src[n,h] = Wh[n,h,:] . a_src[h,:]   s_dst[n,h] = Wh[n,h,:] . a_dst[h,:]
// ---------------------------------------------------------------------------
__global__ void k_scores(const float* __restrict__ Wh,
                         const float* __restrict__ attn,
                         float* __restrict__ ssrc,
                         float* __restrict__ sdst) {
  int i = blockIdx.x * blockDim.x + threadIdx.x;    // i = n*4 + h
  if (i >= N_NODES * N_HEADS) return;
  const int n = i >> 2, h = i & 3;
  const float4* __restrict__ w  = (const float4*)(Wh + n * TOT_OUT + h * OUT_DIM);
  const float4* __restrict__ as = (const float4*)(attn + h * 2 * OUT_DIM);
  const float4* __restrict__ ad = (const float4*)(attn + h * 2 * OUT_DIM + OUT_DIM);
  float s = 0.0f, d = 0.0f;
#pragma unroll
  for (int q = 0; q < OUT_DIM / 4; ++q) {
    float4 wv = w[q], av = as[q], dv = ad[q];
    s += wv.x * av.x + wv.y * av.y + wv.z * av.z + wv.w * av.w;
    d += wv.x * dv.x + wv.y * dv.y + wv.z * dv.z + wv.w * dv.w;
  }
  ssrc[i] = s;
  sdst[i] = d;
}

// ---------------------------------------------------------------------------
// Kernel 3: edge logits + running segment max (float atomic max -> L2)
// ---------------------------------------------------------------------------
__global__ void k_edge_logits(const long long* __restrict__ ei,
                              const float* __restrict__ ssrc,
                              const float* __restrict__ sdst,
                              float* __restrict__ evals,
                              float* __restrict__ maxb) {
  int e = blockIdx.x * blockDim.x + threadIdx.x;
  if (e >= N_EDGES) return;
  const int src = (int)ei[2LL * e];
  const int dst = (int)ei[2LL * e + 1];
#pragma unroll
  for (int h = 0; h < N_HEADS; ++h) {
    float v = ssrc[src * 4 + h] + sdst[dst * 4 + h];
    v = (v > 0.0f) ? v : 0.2f * v;          // leaky_relu(0.2)
    evals[e * 4 + h] = v;
    atomicMax(&maxb[dst * 4 + h], v);
  }
}

// ---------------------------------------------------------------------------
// Kernel 4: nodes with no incoming edges: max stays -inf -> 0 (ref semantics)
// ---------------------------------------------------------------------------
__global__ void k_fix_max(float* __restrict__ maxb) {
  int i = blockIdx.x * blockDim.x + threadIdx.x;
  if (i >= N_NODES * N_HEADS) return;
  float m = maxb[i];
  if (!isfinite(m)) maxb[i] = 0.0f;
}

// ---------------------------------------------------------------------------
// Kernel 5: exp(e - max[dst]) and segment sum
// ---------------------------------------------------------------------------
__global__ void k_edge_exp(const long long* __restrict__ ei,
                           float* __restrict__ evals,
                           const float* __restrict__ maxb,
                           float* __restrict__ sumb) {
  int e = blockIdx.x * blockDim.x + threadIdx.x;
  if (e >= N_EDGES) return;
  const int dst = (int)ei[2LL * e + 1];
#pragma unroll
  for (int h = 0; h < N_HEADS; ++h) {
    float ex = __expf(evals[e * 4 + h] - maxb[dst * 4 + h]);
    evals[e * 4 + h] = ex;
    atomicAdd(&sumb[dst * 4 + h], ex);
  }
}

// ---------------------------------------------------------------------------
// Kernel 6: alpha = exp/sum; out[dst] += alpha * Wh[src]; write alpha.
// One wave32 per edge; lane = out-dim channel within each head.
// Wh and out both fit in L2 (25.6MB each) -> L2-resident gathers/atomics.
// ---------------------------------------------------------------------------
__global__ void k_edge_scatter(const long long* __restrict__ ei,
                               const float* __restrict__ evals,
                               const float* __restrict__ sumb,
                               const float* __restrict__ Wh,
                               float* __restrict__ outh,
                               float* __restrict__ alpha_out) {
  const int gtid = blockIdx.x * blockDim.x + threadIdx.x;
  const int e    = gtid >> 5;
  const int lane = threadIdx.x & 31;
  if (e >= N_EDGES) return;
  const int src = (int)ei[2LL * e];
  const int dst = (int)ei[2LL * e + 1];

  float al[N_HEADS];
#pragma unroll
  for (int h = 0; h < N_HEADS; ++h) {
    al[h] = evals[e * 4 + h] / (sumb[dst * 4 + h] + 1e-9f);
    const float wv = Wh[src * TOT_OUT + h * OUT_DIM + lane];
    atomicAdd(&outh[dst * TOT_OUT + h * OUT_DIM + lane], al[h] * wv);
  }
  if (lane < N_HEADS) alpha_out[e * 4 + lane] = al[lane];
}

// ---------------------------------------------------------------------------
// Workspace layout (bytes, all 256-aligned):
//   Wh    :         0 .. 25,600,000   (50000*128 f32)
//   ssrc  : 25,600,000 + 800,000
//   sdst  : 26,400,000 + 800,000
//   maxb  : 27,200,000 + 800,000
//   sumb  : 28,000,000 + 800,000
//   evals : 28,800,000 + 25,600,000   (1.6M*4 f32)   total 54.4 MB
// ---------------------------------------------------------------------------
extern "C" void kernel_launch(void* const* d_in, const int* in_sizes, int n_in,
                              void* d_out, int out_size, void* d_ws, size_t ws_size,
                              hipStream_t stream) {
  const float*     x      = (const float*)d_in[0];
  const long long* ei     = (const long long*)d_in[1];   // int64 per reference
  const float*     weight = (const float*)d_in[2];
  const float*     attn   = (const float*)d_in[3];

  float* outh  = (float*)d_out;                          // h: 50000*128
  float* alpha = (float*)d_out + (size_t)N_NODES * TOT_OUT;  // alpha: 1.6M*4

  char*  ws    = (char*)d_ws;
  float* Wh    = (float*)(ws);
  float* ssrc  = (float*)(ws + 25600000);
  float* sdst  = (float*)(ws + 26400000);
  float* maxb  = (float*)(ws + 27200000);
  float* sumb  = (float*)(ws + 28000000);
  float* evals = (float*)(ws + 28800000);

  const int B = 256;
  k_init        <<<(N_NODES * TOT_OUT + B - 1) / B, B, 0, stream>>>(outh, sumb, maxb);
  k_gemm_wmma   <<<N_NODES / 16,                     B, 0, stream>>>(x, weight, Wh);
  k_scores      <<<(N_NODES * N_HEADS + B - 1) / B,  B, 0, stream>>>(Wh, attn, ssrc, sdst);
  k_edge_logits <<<(N_EDGES + B - 1) / B,            B, 0, stream>>>(ei, ssrc, sdst, evals, maxb);
  k_fix_max     <<<(N_NODES * N_HEADS + B - 1) / B,  B, 0, stream>>>(maxb);
  k_edge_exp    <<<(N_EDGES + B - 1) / B,            B, 0, stream>>>(ei, evals, maxb, sumb);
  k_edge_scatter<<<(N_EDGES * 32 + B - 1) / B,       B, 0, stream>>>(ei, evals, sumb, Wh, outh, alpha);
}